// Conformer_73864847556659
// MI455X (gfx1250) — compile-verified
//
#include <hip/hip_runtime.h>
#include <math.h>

// ---------------------------------------------------------------------------
// Conformer block, MI455X (gfx1250, wave32).
// F=2 makes Q/K/V rank-3 => MHSA collapses to 3x3 Gram + per-head bias tables
// + softmax. Attention is exp/VALU-bound (134M v_exp_f32), NOT matmul-bound.
// The conv module's GEMMs (pw1 / 31-tap conv / pw2) run on fp32 WMMA
// (v_wmma_f32_16x16x4_f32) to match the fp32 reference exactly on the matrix
// pipe. Total HBM traffic ~5 MB, weights live in L2 (192 MB).
// ---------------------------------------------------------------------------

typedef float v2f __attribute__((ext_vector_type(2)));
typedef float v8f __attribute__((ext_vector_type(8)));

#define BATCH 4
#define CCH   64
#define TT    256
#define FF    2
#define HH    8
#define DD    64
#define DM    512
#define EPS   1e-3f

__device__ __forceinline__ v8f wmma_f32_k4(v2f a, v2f b, v8f c) {
  // D = A(16x4 f32) * B(4x16 f32) + C(16x16 f32)
  return __builtin_amdgcn_wmma_f32_16x16x4_f32(false, a, false, b, (short)0, c,
                                               false, false);
}

// ---------------- FFN (macaron): LN(F=2) -> 2->8 swish -> 8->2, y = x + floor((x+d)/2)
__global__ void k_ffn(const float* __restrict__ x, float* __restrict__ y,
                      const float* __restrict__ g, const float* __restrict__ b,
                      const float* __restrict__ w1, const float* __restrict__ b1,
                      const float* __restrict__ w2, const float* __restrict__ b2,
                      const float* __restrict__ lng, const float* __restrict__ lnb,
                      int final_ln, int n) {
  int i = blockIdx.x * blockDim.x + threadIdx.x;
  if (i >= n) return;
  float x0 = x[2 * i], x1 = x[2 * i + 1];
  float mn = 0.5f * (x0 + x1);
  float dv = x0 - mn;                       // x1 - mn == -dv, var == dv*dv
  float rs = rsqrtf(dv * dv + EPS);
  float n0 = dv * rs * g[0] + b[0];
  float n1 = -dv * rs * g[1] + b[1];
  float d0 = b2[0], d1 = b2[1];
#pragma unroll
  for (int j = 0; j < 8; ++j) {
    float h = n0 * w1[j] + n1 * w1[8 + j] + b1[j];
    h = h / (1.f + expf(-h));               // swish
    d0 = fmaf(h, w2[j * 2 + 0], d0);
    d1 = fmaf(h, w2[j * 2 + 1], d1);
  }
  float o0 = x0 + floorf((x0 + d0) * 0.5f);
  float o1 = x1 + floorf((x1 + d1) * 0.5f);
  if (final_ln) {
    float m2 = 0.5f * (o0 + o1);
    float dd = o0 - m2;
    float r2 = rsqrtf(dd * dd + EPS);
    o0 = dd * r2 * lng[0] + lnb[0];
    o1 = -dd * r2 * lng[1] + lnb[1];
  }
  y[2 * i] = o0;
  y[2 * i + 1] = o1;
}

// ---------------- Attention precompute, part 1 (1 block, 256 threads)
// WDWO[m][f] = wd[m,:]@wo[:,f];  U[c][h][f] = wv[c, h-slice] @ WDWO[h-slice, f]
// Cf[f] = bv@WDWO[:,f] + bd@wo[:,f] + bo[f]
// G[h][a][b] = wq_aug[a, h-slice] . wk_aug[b, h-slice]   (aug row 2 = bias)
// W2[a][h][m] = wq_aug[a, h-slice] . wd[m, h-slice];  Cb[a][h] = wq_aug . bd
__global__ void k_pre1(const float* __restrict__ wq, const float* __restrict__ bq,
                       const float* __restrict__ wk, const float* __restrict__ bk,
                       const float* __restrict__ wv, const float* __restrict__ bv,
                       const float* __restrict__ wd, const float* __restrict__ bd,
                       const float* __restrict__ wo, const float* __restrict__ bo,
                       float* __restrict__ WDWO, float* __restrict__ U,
                       float* __restrict__ Cf, float* __restrict__ G,
                       float* __restrict__ W2, float* __restrict__ Cb) {
  int tid = threadIdx.x;
  for (int e = tid; e < 1024; e += 256) {
    int m = e >> 1, f = e & 1;
    float acc = 0.f;
    for (int j = 0; j < DM; ++j) acc += wd[m * DM + j] * wo[j * 2 + f];
    WDWO[e] = acc;
  }
  for (int e = tid; e < 72; e += 256) {
    int h = e / 9, a = (e % 9) / 3, b2 = e % 3;
    float acc = 0.f;
    for (int d = 0; d < DD; ++d) {
      int c = h * DD + d;
      float qa = (a == 2) ? bq[c] : wq[a * DM + c];
      float kb = (b2 == 2) ? bk[c] : wk[b2 * DM + c];
      acc += qa * kb;
    }
    G[e] = acc;
  }
  for (int e = tid; e < 3 * 8 * DM; e += 256) {
    int a = e / (8 * DM);
    int r = e - a * 8 * DM;
    int h = r >> 9, m = r & 511;
    float acc = 0.f;
    for (int d = 0; d < DD; ++d) {
      int c = h * DD + d;
      float qa = (a == 2) ? bq[c] : wq[a * DM + c];
      acc += qa * wd[m * DM + c];
    }
    W2[e] = acc;
  }
  for (int e = tid; e < 24; e += 256) {
    int a = e >> 3, h = e & 7;
    float acc = 0.f;
    for (int d = 0; d < DD; ++d) {
      int c = h * DD + d;
      float qa = (a == 2) ? bq[c] : wq[a * DM + c];
      acc += qa * bd[c];
    }
    Cb[e] = acc;
  }
  __syncthreads();
  for (int e = tid; e < 32; e += 256) {
    int c = e >> 4, h = (e >> 1) & 7, f = e & 1;
    float acc = 0.f;
    for (int d = 0; d < DD; ++d) {
      int m = h * DD + d;
      acc += wv[c * DM + m] * WDWO[m * 2 + f];
    }
    U[e] = acc;
  }
  if (tid < 2) {
    int f = tid;
    float acc = bo[f];
    for (int m = 0; m < DM; ++m) acc += bv[m] * WDWO[m * 2 + f];
    for (int j = 0; j < DM; ++j) acc += bd[j] * wo[j * 2 + f];
    Cf[f] = acc;
  }
}

// ---------------- Attention precompute, part 2: rel-pos tables
// M[a][h][p] = PE[p,:] @ W2[a][h][:] + Cb[a][h]   (PE generated on the fly)
__global__ void k_pre2(const float* __restrict__ W2, const float* __restrict__ Cb,
                       float* __restrict__ M) {
  __shared__ float pe[DM];
  int p = blockIdx.x;
  for (int m = threadIdx.x; m < DM; m += 256) {
    float expo = -(float)(2 * (m >> 1)) * (9.210340371976184f / (float)DM);
    float ang = (float)p * expf(expo);     // p / 10000^(2*floor(m/2)/DM)
    pe[m] = (m & 1) ? cosf(ang) : sinf(ang);
  }
  __syncthreads();
  if (threadIdx.x < 24) {
    int a = threadIdx.x >> 3, h = threadIdx.x & 7;
    const float* w2 = W2 + (a * 8 + h) * DM;
    float acc = Cb[a * 8 + h];
    for (int m = 0; m < DM; ++m) acc = fmaf(pe[m], w2[m], acc);
    M[(a * 8 + h) * TT + p] = acc;
  }
}

// ---------------- Collapsed MHSA + out-dense + residual.
// logit[s,t] = ([x0s,x1s,1] G [x0t,x1t,1]^T + x0s*M0[t]+x1s*M1[t]+Mb[t]) / 8
__global__ void k_attn(const float* __restrict__ y1, float* __restrict__ y2,
                       const float* __restrict__ g, const float* __restrict__ b,
                       const float* __restrict__ G, const float* __restrict__ M,
                       const float* __restrict__ U, const float* __restrict__ Cf) {
  __shared__ float x0s[TT], x1s[TT], M0[TT], M1[TT], Mb[TT];
  int bc = blockIdx.x;
  int s = threadIdx.x;
  int base = (bc * TT + s) * 2;
  float a0 = y1[base], a1 = y1[base + 1];
  float mn = 0.5f * (a0 + a1);
  float dv = a0 - mn;
  float rs = rsqrtf(dv * dv + EPS);
  x0s[s] = dv * rs * g[0] + b[0];
  x1s[s] = -dv * rs * g[1] + b[1];
  float acc0 = 0.f, acc1 = 0.f;
  for (int h = 0; h < HH; ++h) {
    __syncthreads();
    M0[s] = M[(0 * 8 + h) * TT + s];
    M1[s] = M[(1 * 8 + h) * TT + s];
    Mb[s] = M[(2 * 8 + h) * TT + s];
    __syncthreads();
    float q0 = x0s[s], q1 = x1s[s];
    const float* Gh = G + h * 9;
    float c0 = q0 * Gh[0] + q1 * Gh[3] + Gh[6];
    float c1 = q0 * Gh[1] + q1 * Gh[4] + Gh[7];
    float c2 = q0 * Gh[2] + q1 * Gh[5] + Gh[8];
    float mx = -3.4e38f;
    for (int t = 0; t < TT; ++t) {
      float l = c2 + Mb[t];
      l = fmaf(c0, x0s[t], l);
      l = fmaf(c1, x1s[t], l);
      l = fmaf(q0, M0[t], l);
      l = fmaf(q1, M1[t], l);
      mx = fmaxf(mx, l * 0.125f);
    }
    float sum = 0.f, w0 = 0.f, w1 = 0.f;
    for (int t = 0; t < TT; ++t) {
      float l = c2 + Mb[t];
      l = fmaf(c0, x0s[t], l);
      l = fmaf(c1, x1s[t], l);
      l = fmaf(q0, M0[t], l);
      l = fmaf(q1, M1[t], l);
      float e = __expf(l * 0.125f - mx);
      sum += e;
      w0 = fmaf(e, x0s[t], w0);
      w1 = fmaf(e, x1s[t], w1);
    }
    float inv = 1.f / sum;
    float cc0 = w0 * inv, cc1 = w1 * inv;
    acc0 += cc0 * U[h * 2 + 0] + cc1 * U[16 + h * 2 + 0];
    acc1 += cc0 * U[h * 2 + 1] + cc1 * U[16 + h * 2 + 1];
  }
  y2[base] = a0 + acc0 + Cf[0];
  y2[base + 1] = a1 + acc1 + Cf[1];
}

// ---------------- Conv module helpers
__global__ void k_transpose(const float* __restrict__ y2, float* __restrict__ h1) {
  int idx = blockIdx.x * 256 + threadIdx.x;   // over B*C*T*F = 131072
  int f = idx & 1, t = (idx >> 1) & 255, c = (idx >> 9) & 63, bb = idx >> 15;
  h1[((bb << 8) + t) * 128 + (c << 1) + f] = y2[idx];
}

// Generic fp32 WMMA GEMM wave-tile: C[M,N] = A[M,K] @ B[K,N] + bias (1 wave/tile)
__global__ void k_gemm_pw1(const float* __restrict__ A, const float* __restrict__ Bm,
                           const float* __restrict__ bias, float* __restrict__ C,
                           int M, int N, int K) {
  int wave = threadIdx.x >> 5, lane = threadIdx.x & 31;
  int tile = blockIdx.x * (blockDim.x >> 5) + wave;
  int tilesN = N >> 4;
  int tm = tile / tilesN, tn = tile - tm * tilesN;
  if (tm * 16 >= M) return;                  // uniform per-wave
  int mrow = tm * 16 + (lane & 15);
  int ncol = tn * 16 + (lane & 15);
  int kp = (lane >> 4) << 1;                 // A: VGPR0/1 = K pair, lane halves = K 0/2
  v8f acc = {0.f, 0.f, 0.f, 0.f, 0.f, 0.f, 0.f, 0.f};
  for (int k = 0; k < K; k += 4) {
    v2f a, b;
    a.x = A[mrow * K + k + kp];
    a.y = A[mrow * K + k + kp + 1];
    b.x = Bm[(k + kp) * N + ncol];
    b.y = Bm[(k + kp + 1) * N + ncol];
    acc = wmma_f32_k4(a, b, acc);
  }
  int rbase = tm * 16 + ((lane >> 4) << 3);  // D: VGPR v -> row v / v+8
#pragma unroll
  for (int v = 0; v < 8; ++v) C[(rbase + v) * N + ncol] = acc[v] + bias[ncol];
}

__global__ void k_glu(const float* __restrict__ h2, float* __restrict__ g) {
  int idx = blockIdx.x * 256 + threadIdx.x;  // over 1024*128
  int r = idx >> 7, n = idx & 127;
  float a = h2[r * 256 + n];
  float gt = h2[r * 256 + 128 + n];
  g[idx] = a / (1.f + expf(-gt));
}

// 31-tap 'SAME' conv as K=31*128 WMMA GEMM with shifted A rows (zero halo).
__global__ void k_conv_gemm(const float* __restrict__ A, const float* __restrict__ W,
                            const float* __restrict__ bias, float* __restrict__ C) {
  int wave = threadIdx.x >> 5, lane = threadIdx.x & 31;
  int tile = blockIdx.x * 8 + wave;          // 512 tiles = 64 x 8
  int tm = tile >> 3, tn = tile & 7;
  int mrow = tm * 16 + (lane & 15);
  int bidx = mrow >> 8, trow = mrow & 255;
  int ncol = tn * 16 + (lane & 15);
  int kp = (lane >> 4) << 1;
  v8f acc = {0.f, 0.f, 0.f, 0.f, 0.f, 0.f, 0.f, 0.f};
  for (int w = 0; w < 31; ++w) {
    int ts = trow + w - 15;
    bool ok = (ts >= 0) && (ts < 256);
    const float* Arow = A + (bidx * 256 + ts) * 128;
    const float* Ww = W + w * 16384;
    for (int k = 0; k < 128; k += 4) {
      v2f a, b;
      a.x = ok ? Arow[k + kp] : 0.f;
      a.y = ok ? Arow[k + kp + 1] : 0.f;
      b.x = Ww[(k + kp) * 128 + ncol];
      b.y = Ww[(k + kp + 1) * 128 + ncol];
      acc = wmma_f32_k4(a, b, acc);
    }
  }
  int rbase = tm * 16 + ((lane >> 4) << 3);
#pragma unroll
  for (int v = 0; v < 8; ++v) C[(rbase + v) * 128 + ncol] = acc[v] + bias[ncol];
}

__global__ void k_ln_swish(const float* __restrict__ X, const float* __restrict__ g,
                           const float* __restrict__ b, float* __restrict__ Y) {
  __shared__ float red[128];
  int r = blockIdx.x, t = threadIdx.x;
  float v = X[r * 128 + t];
  red[t] = v;
  __syncthreads();
  for (int s = 64; s > 0; s >>= 1) {
    if (t < s) red[t] += red[t + s];
    __syncthreads();
  }
  float mean = red[0] * (1.f / 128.f);
  __syncthreads();
  float d = v - mean;
  red[t] = d * d;
  __syncthreads();
  for (int s = 64; s > 0; s >>= 1) {
    if (t < s) red[t] += red[t + s];
    __syncthreads();
  }
  float var = red[0] * (1.f / 128.f);
  float xn = d * rsqrtf(var + EPS) * g[t] + b[t];
  Y[r * 128 + t] = xn / (1.f + expf(-xn));   // swish
}

// pw2 GEMM fused with transpose-back + conv-module residual: y3 = y2 + out
__global__ void k_gemm_pw2_res(const float* __restrict__ A, const float* __restrict__ Bm,
                               const float* __restrict__ bias,
                               const float* __restrict__ y2, float* __restrict__ y3) {
  int wave = threadIdx.x >> 5, lane = threadIdx.x & 31;
  int tile = blockIdx.x * 8 + wave;          // 512 tiles = 64 x 8
  int tm = tile >> 3, tn = tile & 7;
  int mrow = tm * 16 + (lane & 15);
  int ncol = tn * 16 + (lane & 15);
  int kp = (lane >> 4) << 1;
  v8f acc = {0.f, 0.f, 0.f, 0.f, 0.f, 0.f, 0.f, 0.f};
  for (int k = 0; k < 128; k += 4) {
    v2f a, b;
    a.x = A[mrow * 128 + k + kp];
    a.y = A[mrow * 128 + k + kp + 1];
    b.x = Bm[(k + kp) * 128 + ncol];
    b.y = Bm[(k + kp + 1) * 128 + ncol];
    acc = wmma_f32_k4(a, b, acc);
  }
  int rbase = tm * 16 + ((lane >> 4) << 3);
  int c = ncol >> 1, f = ncol & 1;
#pragma unroll
  for (int v = 0; v < 8; ++v) {
    int row = rbase + v;
    int bb = row >> 8, t = row & 255;
    int idx = (((bb << 6) + c) * 256 + t) * 2 + f;
    y3[idx] = y2[idx] + acc[v] + bias[ncol];
  }
}

extern "C" void kernel_launch(void* const* d_in, const int* in_sizes, int n_in,
                              void* d_out, int out_size, void* d_ws, size_t ws_size,
                              hipStream_t stream) {
  (void)in_sizes; (void)n_in; (void)out_size; (void)ws_size;
  const float* x      = (const float*)d_in[0];
  const float* ln_f1g = (const float*)d_in[1];  const float* ln_f1b = (const float*)d_in[2];
  const float* w_f1a  = (const float*)d_in[3];  const float* b_f1a  = (const float*)d_in[4];
  const float* w_f1b  = (const float*)d_in[5];  const float* b_f1b  = (const float*)d_in[6];
  const float* ln_mg  = (const float*)d_in[7];  const float* ln_mb  = (const float*)d_in[8];
  const float* wq = (const float*)d_in[9];   const float* bq = (const float*)d_in[10];
  const float* wk = (const float*)d_in[11];  const float* bk = (const float*)d_in[12];
  const float* wv = (const float*)d_in[13];  const float* bv = (const float*)d_in[14];
  const float* wd = (const float*)d_in[15];  const float* bd = (const float*)d_in[16];
  const float* wo = (const float*)d_in[17];  const float* bo = (const float*)d_in[18];
  const float* w_pw1 = (const float*)d_in[19]; const float* b_pw1 = (const float*)d_in[20];
  const float* w_dc  = (const float*)d_in[21]; const float* b_dc  = (const float*)d_in[22];
  const float* ln_cg = (const float*)d_in[23]; const float* ln_cb = (const float*)d_in[24];
  const float* w_pw2 = (const float*)d_in[25]; const float* b_pw2 = (const float*)d_in[26];
  const float* ln_f2g = (const float*)d_in[27]; const float* ln_f2b = (const float*)d_in[28];
  const float* w_f2a  = (const float*)d_in[29]; const float* b_f2a  = (const float*)d_in[30];
  const float* w_f2b  = (const float*)d_in[31]; const float* b_f2b  = (const float*)d_in[32];
  const float* ln_og  = (const float*)d_in[33]; const float* ln_ob  = (const float*)d_in[34];

  float* ws = (float*)d_ws;
  float* y1   = ws;            // 131072
  float* y2   = y1 + 131072;   // 131072
  float* y3   = y2 + 131072;   // 131072
  float* h1   = y3 + 131072;   // 131072
  float* h2   = h1 + 131072;   // 262144
  float* gbuf = h2 + 262144;   // 131072
  float* cvo  = gbuf + 131072; // 131072
  float* sbuf = cvo + 131072;  // 131072
  float* WDWO = sbuf + 131072; // 1024
  float* Ut   = WDWO + 1024;   // 32
  float* Cft  = Ut + 32;       // 2
  float* Gt   = Cft + 2;       // 72
  float* W2t  = Gt + 72;       // 12288
  float* Cbt  = W2t + 12288;   // 24
  float* Mt   = Cbt + 24;      // 6144

  const int NBCT = BATCH * CCH * TT;  // 65536 (b,c,t) rows of F=2

  // 1) macaron FFN1 + floor-residual
  k_ffn<<<NBCT / 256, 256, 0, stream>>>(x, y1, ln_f1g, ln_f1b, w_f1a, b_f1a,
                                        w_f1b, b_f1b, nullptr, nullptr, 0, NBCT);
  // 2) attention precompute (collapsed rank-3 form)
  k_pre1<<<1, 256, 0, stream>>>(wq, bq, wk, bk, wv, bv, wd, bd, wo, bo,
                                WDWO, Ut, Cft, Gt, W2t, Cbt);
  k_pre2<<<TT, 256, 0, stream>>>(W2t, Cbt, Mt);
  // 3) collapsed MHSA + out dense + residual
  k_attn<<<BATCH * CCH, TT, 0, stream>>>(y1, y2, ln_mg, ln_mb, Gt, Mt, Ut, Cft);
  // 4) conv module (fp32 WMMA GEMMs)
  k_transpose<<<512, 256, 0, stream>>>(y2, h1);
  k_gemm_pw1<<<128, 256, 0, stream>>>(h1, w_pw1, b_pw1, h2, 1024, 256, 128);
  k_glu<<<512, 256, 0, stream>>>(h2, gbuf);
  k_conv_gemm<<<64, 256, 0, stream>>>(gbuf, w_dc, b_dc, cvo);
  k_ln_swish<<<1024, 128, 0, stream>>>(cvo, ln_cg, ln_cb, sbuf);
  k_gemm_pw2_res<<<64, 256, 0, stream>>>(sbuf, w_pw2, b_pw2, y2, y3);
  // 5) macaron FFN2 + floor-residual + final LayerNorm -> d_out
  k_ffn<<<NBCT / 256, 256, 0, stream>>>(y3, (float*)d_out, ln_f2g, ln_f2b, w_f2a,
                                        b_f2a, w_f2b, b_f2b, ln_og, ln_ob, 1, NBCT);
}